// Memory_46213848105246
// MI455X (gfx1250) — compile-verified
//
#include <hip/hip_runtime.h>
#include <math.h>

typedef __attribute__((ext_vector_type(2))) float v2f;
typedef __attribute__((ext_vector_type(8))) float v8f;

#define B_TASKS 64
#define D_DIM   512
#define M_SLOTS 8192

// ---------------------------------------------------------------------------
// GEMM 1: S = K (64x512) @ MK0 (512x8192) -> S (64x8192), pure-f32 WMMA.
// One wave per 4 adjacent 16x16 output tiles (A fragment reused 4x);
// K-loop in steps of 4 (V_WMMA_F32_16X16X4_F32).
// ---------------------------------------------------------------------------
__global__ __launch_bounds__(32) void gemm_S_kernel(const float* __restrict__ K,
                                                    const float* __restrict__ MK0,
                                                    float* __restrict__ S) {
  const int lane = threadIdx.x;        // 0..31, one wave
  const int r = lane & 15;
  const int h = lane >> 4;             // 0 or 1
  const int n0 = blockIdx.x * 64;      // 4 column tiles (over M)
  const int row0 = blockIdx.y * 16;    // row tile (over B)
  v8f acc0 = {}, acc1 = {}, acc2 = {}, acc3 = {};
  for (int kk = 0; kk < D_DIM; kk += 4) {
    const int ka = kk + 2 * h;
    v2f a, b;
    a.x = K[(row0 + r) * D_DIM + ka + 0];
    a.y = K[(row0 + r) * D_DIM + ka + 1];
    const float* b0 = &MK0[(ka + 0) * M_SLOTS + n0 + r];
    const float* b1 = &MK0[(ka + 1) * M_SLOTS + n0 + r];
    b.x = b0[0];  b.y = b1[0];
    acc0 = __builtin_amdgcn_wmma_f32_16x16x4_f32(false, a, false, b, (short)0, acc0, false, false);
    b.x = b0[16]; b.y = b1[16];
    acc1 = __builtin_amdgcn_wmma_f32_16x16x4_f32(false, a, false, b, (short)0, acc1, false, false);
    b.x = b0[32]; b.y = b1[32];
    acc2 = __builtin_amdgcn_wmma_f32_16x16x4_f32(false, a, false, b, (short)0, acc2, false, false);
    b.x = b0[48]; b.y = b1[48];
    acc3 = __builtin_amdgcn_wmma_f32_16x16x4_f32(false, a, false, b, (short)0, acc3, false, false);
  }
#pragma unroll
  for (int v = 0; v < 8; ++v) {
    float* row = &S[(row0 + v + 8 * h) * M_SLOTS + n0 + r];
    row[0] = acc0[v]; row[16] = acc1[v]; row[32] = acc2[v]; row[48] = acc3[v];
  }
}

// ---------------------------------------------------------------------------
// GEMM 2: out (64x512) = Wr (64x8192) @ MK0^T (8192x512) + Cuk (64x64) @ K (64x512)
// 4 column tiles per wave.
// ---------------------------------------------------------------------------
__global__ __launch_bounds__(32) void gemm_out_kernel(const float* __restrict__ Wr,
                                                      const float* __restrict__ MK0,
                                                      const float* __restrict__ Cuk,
                                                      const float* __restrict__ K,
                                                      float* __restrict__ out) {
  const int lane = threadIdx.x;
  const int r = lane & 15;
  const int h = lane >> 4;
  const int n0 = blockIdx.x * 64;      // 4 column tiles (over D)
  const int row0 = blockIdx.y * 16;    // row tile (over B)
  v8f acc0 = {}, acc1 = {}, acc2 = {}, acc3 = {};
  // Part 1: Wr @ MK0^T, contraction over M.  B1[k][n] = MK0[n][k]
  for (int kk = 0; kk < M_SLOTS; kk += 4) {
    const int ka = kk + 2 * h;
    v2f a, b;
    a.x = Wr[(row0 + r) * M_SLOTS + ka + 0];
    a.y = Wr[(row0 + r) * M_SLOTS + ka + 1];
    b.x = MK0[(n0 + r + 0) * M_SLOTS + ka + 0];
    b.y = MK0[(n0 + r + 0) * M_SLOTS + ka + 1];
    acc0 = __builtin_amdgcn_wmma_f32_16x16x4_f32(false, a, false, b, (short)0, acc0, false, false);
    b.x = MK0[(n0 + r + 16) * M_SLOTS + ka + 0];
    b.y = MK0[(n0 + r + 16) * M_SLOTS + ka + 1];
    acc1 = __builtin_amdgcn_wmma_f32_16x16x4_f32(false, a, false, b, (short)0, acc1, false, false);
    b.x = MK0[(n0 + r + 32) * M_SLOTS + ka + 0];
    b.y = MK0[(n0 + r + 32) * M_SLOTS + ka + 1];
    acc2 = __builtin_amdgcn_wmma_f32_16x16x4_f32(false, a, false, b, (short)0, acc2, false, false);
    b.x = MK0[(n0 + r + 48) * M_SLOTS + ka + 0];
    b.y = MK0[(n0 + r + 48) * M_SLOTS + ka + 1];
    acc3 = __builtin_amdgcn_wmma_f32_16x16x4_f32(false, a, false, b, (short)0, acc3, false, false);
  }
  // Part 2: Cuk @ K, contraction over B (64)
  for (int kk = 0; kk < B_TASKS; kk += 4) {
    const int ka = kk + 2 * h;
    v2f a, b;
    a.x = Cuk[(row0 + r) * B_TASKS + ka + 0];
    a.y = Cuk[(row0 + r) * B_TASKS + ka + 1];
    const float* b0 = &K[(ka + 0) * D_DIM + n0 + r];
    const float* b1 = &K[(ka + 1) * D_DIM + n0 + r];
    b.x = b0[0];  b.y = b1[0];
    acc0 = __builtin_amdgcn_wmma_f32_16x16x4_f32(false, a, false, b, (short)0, acc0, false, false);
    b.x = b0[16]; b.y = b1[16];
    acc1 = __builtin_amdgcn_wmma_f32_16x16x4_f32(false, a, false, b, (short)0, acc1, false, false);
    b.x = b0[32]; b.y = b1[32];
    acc2 = __builtin_amdgcn_wmma_f32_16x16x4_f32(false, a, false, b, (short)0, acc2, false, false);
    b.x = b0[48]; b.y = b1[48];
    acc3 = __builtin_amdgcn_wmma_f32_16x16x4_f32(false, a, false, b, (short)0, acc3, false, false);
  }
#pragma unroll
  for (int v = 0; v < 8; ++v) {
    float* row = &out[(row0 + v + 8 * h) * D_DIM + n0 + r];
    row[0] = acc0[v]; row[16] = acc1[v]; row[32] = acc2[v]; row[48] = acc3[v];
  }
}

// ---------------------------------------------------------------------------
// Gram matrix G[t][s] = k_t . k_s   (64x64)
// ---------------------------------------------------------------------------
__global__ void gram_kernel(const float* __restrict__ K, float* __restrict__ G) {
  const int idx = blockIdx.x * blockDim.x + threadIdx.x;
  if (idx >= B_TASKS * B_TASKS) return;
  const int t = idx >> 6, s = idx & 63;
  float acc = 0.f;
  for (int d = 0; d < D_DIM; ++d) acc += K[t * D_DIM + d] * K[s * D_DIM + d];
  G[idx] = acc;
}

// ---------------------------------------------------------------------------
// Initial squared column norms of MK0
// ---------------------------------------------------------------------------
__global__ void norms_kernel(const float* __restrict__ MK0, float* __restrict__ n2) {
  const int m = blockIdx.x * blockDim.x + threadIdx.x;
  if (m >= M_SLOTS) return;
  float acc = 0.f;
  for (int d = 0; d < D_DIM; ++d) {
    const float x = MK0[d * M_SLOTS + m];
    acc += x * x;
  }
  n2[m] = acc;
}

// ---------------------------------------------------------------------------
// Block reductions (wave32 + 32-wave block)
// ---------------------------------------------------------------------------
__device__ inline float blockRedSum(float v, float* redF) {
  for (int o = 16; o > 0; o >>= 1) v += __shfl_xor(v, o, 32);
  const int wid = threadIdx.x >> 5, lane = threadIdx.x & 31;
  if (lane == 0) redF[wid] = v;
  __syncthreads();
  if (wid == 0) {
    float x = redF[lane];
    for (int o = 16; o > 0; o >>= 1) x += __shfl_xor(x, o, 32);
    if (lane == 0) redF[0] = x;
  }
  __syncthreads();
  const float r = redF[0];
  __syncthreads();
  return r;
}

__device__ inline float blockRedMax(float v, float* redF) {
  for (int o = 16; o > 0; o >>= 1) v = fmaxf(v, __shfl_xor(v, o, 32));
  const int wid = threadIdx.x >> 5, lane = threadIdx.x & 31;
  if (lane == 0) redF[wid] = v;
  __syncthreads();
  if (wid == 0) {
    float x = redF[lane];
    for (int o = 16; o > 0; o >>= 1) x = fmaxf(x, __shfl_xor(x, o, 32));
    if (lane == 0) redF[0] = x;
  }
  __syncthreads();
  const float r = redF[0];
  __syncthreads();
  return r;
}

__device__ inline void argminPair(float& v, int& i, float ov, int oi) {
  if (ov < v || (ov == v && oi < i)) { v = ov; i = oi; }
}

// First-index argmin across the block (matches jnp.argmin tie-break)
__device__ inline int blockArgmin(float v, int idx, float* redF, int* redI) {
  for (int o = 16; o > 0; o >>= 1) {
    const float ov = __shfl_xor(v, o, 32);
    const int oi = __shfl_xor(idx, o, 32);
    argminPair(v, idx, ov, oi);
  }
  const int wid = threadIdx.x >> 5, lane = threadIdx.x & 31;
  if (lane == 0) { redF[wid] = v; redI[wid] = idx; }
  __syncthreads();
  if (wid == 0) {
    float vv = redF[lane]; int ii = redI[lane];
    for (int o = 16; o > 0; o >>= 1) {
      const float ov = __shfl_xor(vv, o, 32);
      const int oi = __shfl_xor(ii, o, 32);
      argminPair(vv, ii, ov, oi);
    }
    if (lane == 0) redI[0] = ii;
  }
  __syncthreads();
  const int r = redI[0];
  __syncthreads();
  return r;
}

// ---------------------------------------------------------------------------
// Sequential scan: one workgroup (1024 threads = 32 waves), 8 strided columns
// per thread.  M-vectors (w_u, w_w, norm2, last-zero step) live in registers;
// w_r and last-zero are staged in LDS per step so Cuk rows are reduced with
// pure wave32 shuffles (one `s` per wave) instead of ~t block barriers.
// ---------------------------------------------------------------------------
__global__ __launch_bounds__(1024) void scan_kernel(
    const float* __restrict__ S, const float* __restrict__ G,
    const float* __restrict__ norms2_in,
    const float* __restrict__ beta_param, const float* __restrict__ gamma_param,
    float* __restrict__ coef, float* __restrict__ Wr, float* __restrict__ Cuk) {
  __shared__ float ldsG[64];
  __shared__ float redF[32];
  __shared__ int redI[32];
  __shared__ float ldsWr[M_SLOTS];        // 32 KB
  __shared__ signed char ldsLz[M_SLOTS];  // 8 KB (values in [-1, 63])

  const int tid = threadIdx.x;
  const float beta = 1.f / (1.f + __expf(-beta_param[0]));
  const float gamma = gamma_param[0];
  const float mean_wr = 1.0f / (float)M_SLOTS;  // softmax sums to 1

  float wu[8], ww[8], n2[8];
  int lz[8];  // last step this column was zeroed (-1 = never)
#pragma unroll
  for (int i = 0; i < 8; ++i) {
    wu[i] = 0.f; ww[i] = 0.f; lz[i] = -1;
    n2[i] = norms2_in[tid + i * 1024];
  }
  int lu_prev = -1;  // previous one-hot least-used index (-1 = zeros vector)

  for (int t = 0; t < B_TASKS; ++t) {
    if (tid < 64) ldsG[tid] = G[t * 64 + tid];
    __syncthreads();
    const float nk2 = ldsG[t];
    const float rnk = rsqrtf(nk2);
    if (t + 1 < B_TASKS) {
      __builtin_prefetch(&S[(t + 1) * M_SLOTS + tid], 0, 0);
      if (tid < 64) __builtin_prefetch(&G[(t + 1) * 64 + tid], 0, 0);
    }

    // dot_t[m] = alive(MK0)*S[t,m] + sum_{s in [lz, t)} G[t,s]*coef[s][m]
    float dot[8];
#pragma unroll
    for (int i = 0; i < 8; ++i) {
      const int m = tid + i * 1024;
      float d = (lz[i] < 0) ? S[t * M_SLOTS + m] : 0.f;
      for (int s = 0; s < t; ++s) {
        const float c = coef[s * M_SLOTS + m];
        d += (s >= lz[i]) ? ldsG[s] * c : 0.f;
      }
      dot[i] = d;
    }

    // cosine similarity + softmax
    float sim[8], lmax = -3.4e38f;
#pragma unroll
    for (int i = 0; i < 8; ++i) {
      sim[i] = dot[i] * rnk * rsqrtf(fmaxf(n2[i], 1e-30f));
      lmax = fmaxf(lmax, sim[i]);
    }
    const float smax = blockRedMax(lmax, redF);
    float wr[8], lsum = 0.f;
#pragma unroll
    for (int i = 0; i < 8; ++i) { wr[i] = __expf(sim[i] - smax); lsum += wr[i]; }
    const float ssum = blockRedSum(lsum, redF);
    const float inv = 1.f / ssum;
#pragma unroll
    for (int i = 0; i < 8; ++i) wr[i] *= inv;

    // argmin of w_u BEFORE update -> column to zero this step
    float av = 3.4e38f; int ai = 0x7fffffff;
#pragma unroll
    for (int i = 0; i < 8; ++i)
      if (wu[i] < av) { av = wu[i]; ai = tid + i * 1024; }
    const int j3 = blockArgmin(av, ai, redF, redI);

    // w_u update uses OLD w_w
#pragma unroll
    for (int i = 0; i < 8; ++i) wu[i] = gamma * wu[i] + wr[i] + ww[i];

    // new write weights: beta*mean(w_r) + (1-beta)*w_lu_prev
    float wwn[8];
    const float base = beta * mean_wr;
#pragma unroll
    for (int i = 0; i < 8; ++i) {
      const int m = tid + i * 1024;
      wwn[i] = base + (1.f - beta) * ((m == lu_prev) ? 1.f : 0.f);
    }

    // argmin of UPDATED w_u -> next one-hot least-used
    av = 3.4e38f; ai = 0x7fffffff;
#pragma unroll
    for (int i = 0; i < 8; ++i)
      if (wu[i] < av) { av = wu[i]; ai = tid + i * 1024; }
    lu_prev = blockArgmin(av, ai, redF, redI);

    // apply step-t zeroing (happens before the u_ref read)
#pragma unroll
    for (int i = 0; i < 8; ++i)
      if (tid + i * 1024 == j3) lz[i] = t;

    // record adjusted read weights / write coefficients; stage wr, lz in LDS
#pragma unroll
    for (int i = 0; i < 8; ++i) {
      const int m = tid + i * 1024;
      Wr[t * M_SLOTS + m] = (lz[i] < 0) ? wr[i] : 0.f;
      coef[t * M_SLOTS + m] = wwn[i];
      ldsWr[m] = wr[i];
      ldsLz[m] = (signed char)lz[i];
    }
    __syncthreads();

    // Cuk[t][s] = sum_m w_r[m]*coef[s][m]*alive_after_zero  (s<t), else 0.
    // Each wave owns s = wid, wid+32; pure wave-level shuffle reduction.
    {
      const int wid = tid >> 5, lane = tid & 31;
      for (int s = wid; s < B_TASKS; s += 32) {
        if (s < t) {
          float p = 0.f;
          for (int m = lane; m < M_SLOTS; m += 32) {
            const float c = coef[s * M_SLOTS + m];
            p += (s >= (int)ldsLz[m]) ? ldsWr[m] * c : 0.f;
          }
          for (int o = 16; o > 0; o >>= 1) p += __shfl_xor(p, o, 32);
          if (lane == 0) Cuk[t * 64 + s] = p;
        } else if (lane == 0) {
          Cuk[t * 64 + s] = 0.f;
        }
      }
    }

    // incremental column-norm update for next step
#pragma unroll
    for (int i = 0; i < 8; ++i) {
      const int m = tid + i * 1024;
      const float dz = (m == j3) ? 0.f : dot[i];
      const float nz = (m == j3) ? 0.f : n2[i];
      n2[i] = nz + 2.f * wwn[i] * dz + wwn[i] * wwn[i] * nk2;
      ww[i] = wwn[i];
    }
    __syncthreads();  // protects ldsWr/ldsLz/ldsG reuse next step
  }
}

// ---------------------------------------------------------------------------
extern "C" void kernel_launch(void* const* d_in, const int* in_sizes, int n_in,
                              void* d_out, int out_size, void* d_ws, size_t ws_size,
                              hipStream_t stream) {
  const float* K   = (const float*)d_in[0];  // [64,1,512]
  // d_in[1] (u) and d_in[3] (memory_understanding) are dead w.r.t. the output
  const float* MK0 = (const float*)d_in[2];  // [512,8192]
  const float* bet = (const float*)d_in[4];
  const float* gam = (const float*)d_in[5];
  float* out = (float*)d_out;                // [64,1,512]

  float* ws = (float*)d_ws;
  float* S      = ws;                         // 64*8192
  float* coef   = S + 64 * 8192;              // 64*8192
  float* Wr     = coef + 64 * 8192;           // 64*8192
  float* G      = Wr + 64 * 8192;             // 64*64
  float* Cuk    = G + 64 * 64;                // 64*64
  float* norms2 = Cuk + 64 * 64;              // 8192

  gemm_S_kernel<<<dim3(M_SLOTS / 64, B_TASKS / 16), 32, 0, stream>>>(K, MK0, S);
  gram_kernel<<<(B_TASKS * B_TASKS + 255) / 256, 256, 0, stream>>>(K, G);
  norms_kernel<<<M_SLOTS / 256, 256, 0, stream>>>(MK0, norms2);
  scan_kernel<<<1, 1024, 0, stream>>>(S, G, norms2, bet, gam, coef, Wr, Cuk);
  gemm_out_kernel<<<dim3(D_DIM / 64, B_TASKS / 16), 32, 0, stream>>>(Wr, MK0, Cuk, K, out);
}